// DynamicDiffeomorphism_8143257993681
// MI455X (gfx1250) — compile-verified
//
#include <hip/hip_runtime.h>
#include <math.h>

typedef __attribute__((ext_vector_type(2))) float v2f;
typedef __attribute__((ext_vector_type(8))) float v8f;

#define NPIX 512
#define NBATCH 64
#define NCHAN 3
#define NCUT 4

// One wave (32 lanes) computes one 16x16 (y,x) tile of one batch:
//   1) builds A = s_y (16x4) and B = Tx (4x16) fragments in registers,
//   2) two V_WMMA_F32_16X16X4_F32 -> u,v field tiles (f32, exact K=4 GEMM),
//   3) converts to displacements and does the fused bilinear gather for 3 channels.
__global__ __launch_bounds__(128) void diffeo_fused_kernel(
    const float* __restrict__ img,   // [B, C, N, N]
    const float* __restrict__ Fx,    // [B, 4, 4]
    const float* __restrict__ Fy,    // [B, 4, 4]
    float* __restrict__ out)         // [B, C, N, N]
{
    const int lane   = threadIdx.x & 31;
    const int waveId = threadIdx.x >> 5;              // 0..3
    const int tile   = blockIdx.x * 4 + waveId;       // 0..1023
    const int b      = blockIdx.y;                    // batch
    const int ty     = tile >> 5;                     // 0..31
    const int tx     = tile & 31;                     // 0..31
    const int y0     = ty * 16;
    const int x0     = tx * 16;

    const int half = lane >> 4;      // selects K-pair {0,1} vs {2,3}
    const int l15  = lane & 15;
    const int kb   = half * 2;

    const float PI  = 3.14159265358979323846f;
    const float inv = 1.0f / (float)(NPIX - 1);

    // scale = sqrt(T_TEMP / typ^2) * n, typ = n*sqrt(pi*log(cut))/2  (const-folds)
    const float typ   = (float)NPIX * sqrtf(PI * logf((float)NCUT)) * 0.5f;
    const float scale = 0.1f * (float)NPIX / typ;

    // ---- A fragment: A[m, j] = sin(pi * (y0+m)/(n-1) * (j+1)) ----
    const float yarg = PI * (float)(y0 + l15) * inv;
    v2f afrag;
    afrag.x = __sinf(yarg * (float)(kb + 1));
    afrag.y = __sinf(yarg * (float)(kb + 2));

    // ---- B fragments: B[j, nn] = Tx(x0+nn, j) = sum_i Fe[i,j] * s_x(x0+nn, i) ----
    const float xarg = PI * (float)(x0 + l15) * inv;
    float sx0 = __sinf(xarg * 1.0f);
    float sx1 = __sinf(xarg * 2.0f);
    float sx2 = __sinf(xarg * 3.0f);
    float sx3 = __sinf(xarg * 4.0f);

    const float* fxb = Fx + (size_t)b * 16;
    const float* fyb = Fy + (size_t)b * 16;

    v2f bfx, bfy;
    #pragma unroll
    for (int jj = 0; jj < 2; ++jj) {
        const int j = kb + jj;
        float tx_acc = 0.0f, ty_acc = 0.0f;
        #pragma unroll
        for (int i = 0; i < 4; ++i) {
            // radial cutoff envelope: e = (r < cut+0.5) / r, r = sqrt((i+1)^2+(j+1)^2)
            const float ki = (float)(i + 1), kj = (float)(j + 1);
            const float r  = sqrtf(ki * ki + kj * kj);
            const float e  = (r < (float)NCUT + 0.5f) ? (1.0f / r) : 0.0f;
            const float si = (i == 0) ? sx0 : (i == 1) ? sx1 : (i == 2) ? sx2 : sx3;
            tx_acc = fmaf(fxb[i * 4 + j] * e, si, tx_acc);
            ty_acc = fmaf(fyb[i * 4 + j] * e, si, ty_acc);
        }
        if (jj == 0) { bfx.x = tx_acc; bfy.x = ty_acc; }
        else         { bfx.y = tx_acc; bfy.y = ty_acc; }
    }

    // ---- WMMA: u = A x Bx, v = A x By (16x16 f32 tiles, K=4) ----
    v8f u = {};
    v8f v = {};
    u = __builtin_amdgcn_wmma_f32_16x16x4_f32(false, afrag, false, bfx,
                                              (short)0, u, false, false);
    v = __builtin_amdgcn_wmma_f32_16x16x4_f32(false, afrag, false, bfy,
                                              (short)0, v, false, false);

    // ---- fused bilinear gather: each lane owns 8 sample points (col fixed) ----
    const int x_col = x0 + l15;
    const float fN1 = (float)(NPIX - 1);
    const size_t plane = (size_t)NPIX * NPIX;
    const float* imgb = img + (size_t)b * NCHAN * plane;
    float*       outb = out + (size_t)b * NCHAN * plane;

    #pragma unroll
    for (int r = 0; r < 8; ++r) {
        const int y = y0 + r + half * 8;       // C/D layout: M = r (+8 upper half)
        const float dx = scale * u[r];
        const float dy = scale * v[r];

        float xn = fminf(fmaxf((float)x_col - dx, 0.0f), fN1);
        float yn = fminf(fmaxf((float)y     - dy, 0.0f), fN1);

        const float xff = floorf(xn), yff = floorf(yn);
        const int xf = (int)xff,        yf = (int)yff;
        const int xc = (int)ceilf(xn),  yc = (int)ceilf(yn);
        const float wx = xn - xff,      wy = yn - yff;

        const float w00 = (1.0f - wy) * (1.0f - wx);
        const float w01 = (1.0f - wy) * wx;
        const float w10 = wy * (1.0f - wx);
        const float w11 = wy * wx;

        const int o00 = yf * NPIX + xf;
        const int o01 = yf * NPIX + xc;
        const int o10 = yc * NPIX + xf;
        const int o11 = yc * NPIX + xc;
        const size_t oidx = (size_t)y * NPIX + x_col;

        #pragma unroll
        for (int c = 0; c < NCHAN; ++c) {
            const float* p = imgb + (size_t)c * plane;
            const float v00 = p[o00];
            const float v01 = p[o01];
            const float v10 = p[o10];
            const float v11 = p[o11];
            outb[(size_t)c * plane + oidx] =
                w00 * v00 + w01 * v01 + w10 * v10 + w11 * v11;
        }
    }
}

extern "C" void kernel_launch(void* const* d_in, const int* in_sizes, int n_in,
                              void* d_out, int out_size, void* d_ws, size_t ws_size,
                              hipStream_t stream) {
    const float* x  = (const float*)d_in[0];
    const float* Fx = (const float*)d_in[1];
    const float* Fy = (const float*)d_in[2];
    float* out = (float*)d_out;

    // 1024 tiles of 16x16 per batch image; 4 waves (tiles) per 128-thread block.
    dim3 grid(256, NBATCH, 1);
    dim3 block(128, 1, 1);
    hipLaunchKernelGGL(diffeo_fused_kernel, grid, block, 0, stream, x, Fx, Fy, out);
}